// Top2Gating_609885356784
// MI455X (gfx1250) — compile-verified
//
#include <hip/hip_runtime.h>

typedef float v2f __attribute__((ext_vector_type(2)));
typedef float v8f __attribute__((ext_vector_type(8)));

constexpr int B = 8, N = 2048, D = 1024, E = 64, CAP = 64;
constexpr int T = B * N;                       // 16384 tokens
constexpr long long OUT_COMBINE = (long long)T * E * CAP;   // dispatch first, combine second
constexpr long long OUT_LOSS    = 2LL * T * E * CAP;        // scalar loss at the end

// ---------------------------------------------------------------------------
// Kernel 1: logits = x @ W  via V_WMMA_F32_16X16X4_F32
// block = 256 threads (8 waves). Each wave computes a 16-row x 64-expert strip.
// K tiled by 64 through LDS. W stored TRANSPOSED in LDS so B operands are a
// single aligned ds_load_b64 (k, k+1 adjacent), matching the A-operand path.
// ---------------------------------------------------------------------------
__global__ __launch_bounds__(256) void gemm_logits(const float* __restrict__ x,
                                                   const float* __restrict__ Wm,
                                                   float* __restrict__ logits) {
    __shared__ float sWt[64][66];       // [expert][k], pad 2 keeps b64 alignment
    __shared__ float sA[8][16][68];     // [wave][row][k], stride 272B = 17*16B

    const int tid  = threadIdx.x;
    const int wave = tid >> 5;
    const int lane = tid & 31;
    const int l16  = lane & 15;
    const int kp   = (lane >> 4) << 1;          // lanes 0-15: k+0/+1, lanes 16-31: k+2/+3
    const int blockRow = blockIdx.x * 128;
    const int rowBase  = blockRow + wave * 16;

    v8f acc0 = {}, acc1 = {}, acc2 = {}, acc3 = {};

    for (int kc = 0; kc < D; kc += 64) {
        // stage W chunk transposed: 64 k-rows x 64 experts, float4 global reads
        for (int i = tid; i < 64 * 16; i += 256) {
            const int kk = i >> 4;
            const int c4 = (i & 15) << 2;
            const float4 w = *(const float4*)&Wm[(long long)(kc + kk) * E + c4];
            sWt[c4 + 0][kk] = w.x;
            sWt[c4 + 1][kk] = w.y;
            sWt[c4 + 2][kk] = w.z;
            sWt[c4 + 3][kk] = w.w;
        }
        // stage A chunk: 128 rows x 64 k, float4 global reads + float4 LDS stores
        for (int i = tid; i < 128 * 16; i += 256) {
            const int r  = i >> 4;
            const int k4 = (i & 15) << 2;
            const float4 a = *(const float4*)&x[(long long)(blockRow + r) * D + kc + k4];
            *(float4*)&sA[r >> 4][r & 15][k4] = a;
        }
        __syncthreads();

        #pragma unroll
        for (int kk = 0; kk < 64; kk += 4) {
            const v2f a  = *(const v2f*)&sA[wave][l16][kk + kp];
            const v2f b0 = *(const v2f*)&sWt[l16][kk + kp];
            const v2f b1 = *(const v2f*)&sWt[l16 + 16][kk + kp];
            const v2f b2 = *(const v2f*)&sWt[l16 + 32][kk + kp];
            const v2f b3 = *(const v2f*)&sWt[l16 + 48][kk + kp];
            acc0 = __builtin_amdgcn_wmma_f32_16x16x4_f32(false, a, false, b0, (short)0, acc0, false, false);
            acc1 = __builtin_amdgcn_wmma_f32_16x16x4_f32(false, a, false, b1, (short)0, acc1, false, false);
            acc2 = __builtin_amdgcn_wmma_f32_16x16x4_f32(false, a, false, b2, (short)0, acc2, false, false);
            acc3 = __builtin_amdgcn_wmma_f32_16x16x4_f32(false, a, false, b3, (short)0, acc3, false, false);
        }
        __syncthreads();
    }

    // C layout: VGPR r, lanes 0-15 -> M=r,N=lane ; lanes 16-31 -> M=8+r,N=lane-16
    const int mhi = (lane >> 4) << 3;
    #pragma unroll
    for (int r = 0; r < 8; ++r) {
        long long rowOff = (long long)(rowBase + r + mhi) * E;
        logits[rowOff + l16]      = acc0[r];
        logits[rowOff + l16 + 16] = acc1[r];
        logits[rowOff + l16 + 32] = acc2[r];
        logits[rowOff + l16 + 48] = acc3[r];
    }
}

// ---------------------------------------------------------------------------
// Kernel 2: per-token softmax over 64 experts + top-2 + loss statistics
// one thread per token; blocks align with batch boundaries (2048 % 256 == 0)
// ---------------------------------------------------------------------------
__global__ __launch_bounds__(256) void softmax_top2(const float* __restrict__ logits,
                                                    float* __restrict__ g1o,
                                                    float* __restrict__ g2o,
                                                    int* __restrict__ meta,
                                                    float* __restrict__ proxy,
                                                    int* __restrict__ cnt) {
    const int t = blockIdx.x * 256 + threadIdx.x;
    const int b = t >> 11;
    const float4* row = reinterpret_cast<const float4*>(logits + (long long)t * E);

    float mx = -1e30f;
    #pragma unroll
    for (int i = 0; i < 16; ++i) {
        float4 v = row[i];
        mx = fmaxf(mx, fmaxf(fmaxf(v.x, v.y), fmaxf(v.z, v.w)));
    }
    float s = 0.f;
    #pragma unroll
    for (int i = 0; i < 16; ++i) {
        float4 v = row[i];
        s += __expf(v.x - mx) + __expf(v.y - mx) + __expf(v.z - mx) + __expf(v.w - mx);
    }
    const float inv = 1.f / s;

    float g1 = -1.f, g2 = -1.f;
    int i1 = 0, i2 = 0;
    #pragma unroll
    for (int i = 0; i < 16; ++i) {
        float4 v = row[i];
        float r4[4] = { __expf(v.x - mx) * inv, __expf(v.y - mx) * inv,
                        __expf(v.z - mx) * inv, __expf(v.w - mx) * inv };
        #pragma unroll
        for (int j = 0; j < 4; ++j) {
            const int e = i * 4 + j;
            const float r = r4[j];
            // wave-level reduction of raw_gates for density_1_proxy (wave32)
            float ps = r;
            for (int off = 16; off > 0; off >>= 1) ps += __shfl_down(ps, off, 32);
            if ((threadIdx.x & 31) == 0) atomicAdd(&proxy[b * E + e], ps);
            if (r > g1)      { g2 = g1; i2 = i1; g1 = r; i1 = e; }
            else if (r > g2) { g2 = r; i2 = e; }
        }
    }
    const float denom = g1 + g2 + 1e-9f;
    const float g1n = g1 / denom;
    const float g2n = g2 / denom;
    const int flag2 = (g2n > 0.2f) ? 1 : 0;      // threshold policy

    g1o[t] = g1n;
    g2o[t] = g2n;
    meta[t] = i1 | (i2 << 8) | (flag2 << 16);
    atomicAdd(&cnt[b * E + i1], 1);              // density_1 (pre-capacity top1 counts)
}

// ---------------------------------------------------------------------------
// Kernel 3: per-batch exclusive cumsum over tokens -> positions in experts
// one workgroup per batch; 8 chunks of 256 tokens, sequential across chunks
// ---------------------------------------------------------------------------
__global__ __launch_bounds__(256) void scan_positions(const int* __restrict__ meta,
                                                      int* __restrict__ posm) {
    __shared__ int s_idx[256];
    __shared__ int run[64];
    __shared__ int base2[64];
    const int tid = threadIdx.x;
    const int tokBase = blockIdx.x * N;

    int idx1[8], idx2[8], flag2[8], r1[8], r2[8];

    if (tid < 64) run[tid] = 0;
    __syncthreads();

    // pass 1: top-1 ranks
    for (int c = 0; c < 8; ++c) {
        const int t = tokBase + c * 256 + tid;
        const int m = meta[t];
        idx1[c]  = m & 255;
        idx2[c]  = (m >> 8) & 255;
        flag2[c] = (m >> 16) & 1;
        s_idx[tid] = idx1[c];
        __syncthreads();
        int r = run[idx1[c]];
        for (int j = 0; j < tid; ++j) r += (s_idx[j] == idx1[c]) ? 1 : 0;
        r1[c] = r;
        __syncthreads();
        if (tid < 64) {
            int cadd = 0;
            for (int j = 0; j < 256; ++j) cadd += (s_idx[j] == tid) ? 1 : 0;
            run[tid] += cadd;
        }
        __syncthreads();
    }

    // mask_1_count = min(total_top1_count, CAP); reset running counts
    if (tid < 64) base2[tid] = min(run[tid], CAP);
    __syncthreads();
    if (tid < 64) run[tid] = 0;
    __syncthreads();

    // pass 2: top-2 ranks (only thresholded tokens participate)
    for (int c = 0; c < 8; ++c) {
        s_idx[tid] = flag2[c] ? idx2[c] : 255;
        __syncthreads();
        int r = 0;
        if (flag2[c]) {
            r = run[idx2[c]];
            for (int j = 0; j < tid; ++j) r += (s_idx[j] == idx2[c]) ? 1 : 0;
            r += base2[idx2[c]];
        }
        r2[c] = r;
        __syncthreads();
        if (tid < 64) {
            int cadd = 0;
            for (int j = 0; j < 256; ++j) cadd += (s_idx[j] == tid) ? 1 : 0;
            run[tid] += cadd;
        }
        __syncthreads();
    }

    for (int c = 0; c < 8; ++c) {
        const int t = tokBase + c * 256 + tid;
        const int keep1 = (r1[c] < CAP) ? 1 : 0;
        const int keep2 = (flag2[c] && r2[c] < CAP) ? 1 : 0;
        posm[t] = (r1[c] & 63) | ((r2[c] & 63) << 6) | (keep1 << 12) | (keep2 << 13);
    }
}

// ---------------------------------------------------------------------------
// Kernel 4: vectorized zero fill of dispatch+combine (the 537 MB store wall)
// ---------------------------------------------------------------------------
__global__ void zero_out(float4* __restrict__ out, long long n4) {
    long long i = (long long)blockIdx.x * blockDim.x + threadIdx.x;
    const long long stride = (long long)gridDim.x * blockDim.x;
    const float4 z = {0.f, 0.f, 0.f, 0.f};
    for (; i < n4; i += stride) out[i] = z;
}

// ---------------------------------------------------------------------------
// Kernel 5: scatter <=2 nonzeros per token into dispatch/combine
// ---------------------------------------------------------------------------
__global__ __launch_bounds__(256) void scatter(const float* __restrict__ g1o,
                                               const float* __restrict__ g2o,
                                               const int* __restrict__ meta,
                                               const int* __restrict__ posm,
                                               float* __restrict__ out) {
    const int t = blockIdx.x * 256 + threadIdx.x;
    const int m = meta[t];
    const int p = posm[t];
    const long long dbase = (long long)t * (E * CAP);
    const long long cbase = OUT_COMBINE + dbase;
    if (p & (1 << 12)) {
        const int e1 = m & 255, pos1 = p & 63;
        const long long off = (long long)e1 * CAP + pos1;
        out[cbase + off] = g1o[t];
        out[dbase + off] = 1.0f;
    }
    if (p & (1 << 13)) {
        const int e2 = (m >> 8) & 255, pos2 = (p >> 6) & 63;
        const long long off = (long long)e2 * CAP + pos2;
        out[cbase + off] = g2o[t];
        out[dbase + off] = 1.0f;
    }
}

// ---------------------------------------------------------------------------
// Kernel 6: loss = mean_{b,e}(proxy_mean * density) * E^2
// ---------------------------------------------------------------------------
__global__ __launch_bounds__(512) void loss_kernel(const float* __restrict__ proxy,
                                                   const int* __restrict__ cnt,
                                                   float* __restrict__ out) {
    __shared__ float red[512];
    const int i = threadIdx.x;
    red[i] = (proxy[i] * (1.f / N)) * ((float)cnt[i] * (1.f / N));
    __syncthreads();
    for (int s = 256; s > 0; s >>= 1) {
        if (i < s) red[i] += red[i + s];
        __syncthreads();
    }
    if (i == 0) out[OUT_LOSS] = red[0] * ((float)(E * E) / (float)(B * E));
}

__global__ void init_accum(float* __restrict__ proxy, int* __restrict__ cnt) {
    const int i = blockIdx.x * blockDim.x + threadIdx.x;
    if (i < B * E) { proxy[i] = 0.f; cnt[i] = 0; }
}

// ---------------------------------------------------------------------------
extern "C" void kernel_launch(void* const* d_in, const int* in_sizes, int n_in,
                              void* d_out, int out_size, void* d_ws, size_t ws_size,
                              hipStream_t stream) {
    const float* x  = (const float*)d_in[0];
    const float* Wm = (const float*)d_in[1];
    float* out = (float*)d_out;

    // workspace layout (words)
    float* logits = (float*)d_ws;                       // T*E
    float* g1o    = logits + (long long)T * E;          // T
    float* g2o    = g1o + T;                            // T
    int*   meta   = (int*)(g2o + T);                    // T
    int*   posm   = meta + T;                           // T
    float* proxy  = (float*)(posm + T);                 // B*E
    int*   cnt    = (int*)(proxy + B * E);              // B*E

    hipLaunchKernelGGL(init_accum, dim3(2), dim3(256), 0, stream, proxy, cnt);
    hipLaunchKernelGGL(zero_out, dim3(16384), dim3(256), 0, stream,
                       (float4*)out, (long long)(2LL * T * E * CAP / 4));
    hipLaunchKernelGGL(gemm_logits, dim3(T / 128), dim3(256), 0, stream, x, Wm, logits);
    hipLaunchKernelGGL(softmax_top2, dim3(T / 256), dim3(256), 0, stream,
                       logits, g1o, g2o, meta, proxy, cnt);
    hipLaunchKernelGGL(scan_positions, dim3(B), dim3(256), 0, stream, meta, posm);
    hipLaunchKernelGGL(scatter, dim3(T / 256), dim3(256), 0, stream, g1o, g2o, meta, posm, out);
    hipLaunchKernelGGL(loss_kernel, dim3(1), dim3(512), 0, stream, proxy, cnt, out);
}